// Sac4DplWithNnModule_21363167330828
// MI455X (gfx1250) — compile-verified
//
#include <hip/hip_runtime.h>
#include <stdint.h>

// SAC-SMA recurrence: one basin per lane, sequential over T.
// CDNA5 path: deep prefetch pipeline via global_load_async_to_lds_b64 +
// s_wait_asynccnt (ASYNCcnt), consume from LDS with ds_load_b64.

#define BLOCK 64
#define CHUNK 16   // timesteps per prefetch chunk
#define DEPTH 3    // chunks in flight (ring buffer in LDS)

__device__ __constant__ float c_lo[21] = {
    0.1f, 0.0f, 0.0f, 1.0f, 1.0f, 1.0f, 1.0f, 1.0f, 0.0f, 0.0f, 0.0f,
    1.0f, 0.0f, 0.1f, 0.01f, 0.001f, 0.0f, 0.98f, 0.99f, 0.0f, 0.0f};
__device__ __constant__ float c_hi[21] = {
    1.0f, 0.1f, 0.3f, 150.0f, 150.0f, 500.0f, 1000.0f, 1000.0f, 0.3f, 0.6f, 0.3f,
    250.0f, 5.0f, 0.5f, 0.35f, 0.05f, 0.9f, 0.998f, 0.999f, 1.0f, 0.5f};

// per-lane 8-byte async copy: copies from global address base + goff to LDS offset lds_off  (GVS mode)
__device__ __forceinline__ void async_copy_f2(uint32_t lds_off, const float* base,
                                              uint32_t goff) {
    asm volatile("global_load_async_to_lds_b64 %0, %1, %2"
                 :: "v"(lds_off), "v"(goff), "s"(base)
                 : "memory");
}

__global__ __launch_bounds__(BLOCK) void sac_sma_kernel(
    const float* __restrict__ pe,    // [T, B, 2]
    const float* __restrict__ prm,   // [B, 21] normalized
    float* __restrict__ out,         // [T, B, 2] (q_sim, e_sim)
    int T, int B)
{
    __shared__ float2 buf[DEPTH * CHUNK * BLOCK];

    const int tid = threadIdx.x;
    const int b  = blockIdx.x * BLOCK + tid;
    const int bc = (b < B) ? b : (B - 1);
    const bool live = (b < B);

    // ---- scale parameters to physical ranges ----
    float P[21];
#pragma unroll
    for (int i = 0; i < 21; ++i)
        P[i] = c_lo[i] + prm[(size_t)bc * 21 + i] * (c_hi[i] - c_lo[i]);

    const float kc = P[0],  pctim = P[1],  adimp = P[2],  uztwm = P[3];
    const float uzfwm = P[4], lztwm = P[5], lzfsm = P[6], lzfpm = P[7];
    const float rserv = P[8], pfree = P[9], riva = P[10], zperc = P[11];
    const float rexp = P[12], uzk = P[13], lzsk = P[14], lzpk = P[15];
    const float ci = P[16], cgs = P[17], cgp = P[18];   // ke, xe unused

    // ---- per-basin invariants ----
    const float parea   = 1.0f - pctim - adimp;
    const float pbase   = lzfsm * lzsk + lzfpm * lzpk;
    const float u       = parea * 1000000.0f;            // parea * AREA * 1000
    const float lzfsum  = lzfsm + lzfpm;
    const float saved   = rserv * lzfsum;
    const float inv_uztwm = 1.0f / uztwm;
    const float inv_lztwm = 1.0f / lztwm;
    const float inv_ul    = 1.0f / (uztwm + lztwm);
    const float inv_uzfwm = 1.0f / uzfwm;
    const float cap_l     = lzfsum + lztwm;
    const float inv_cap_l = 1.0f / cap_l;
    const float inv_uu    = 1.0f / (uztwm + uzfwm);
    const float inv_rden  = 1.0f / (lzfsum - saved + lztwm);
    const float inv_lzfpm = 1.0f / lzfpm;
    const float coef_a    = lzfpm / lzfsum;
    const float kiu  = (1.0f - ci)  * u;
    const float kgsu = (1.0f - cgs) * u;
    const float kgpu = (1.0f - cgp) * u;

    const uint32_t lds0 = (uint32_t)(uintptr_t)(&buf[0]);   // low 32 bits = LDS offset
    const int nchunk = (T + CHUNK - 1) / CHUNK;

    // ---- prologue: fill the async pipeline ----
    for (int c = 0; c < DEPTH && c < nchunk; ++c) {
        const uint32_t sb = lds0 + (uint32_t)(c * CHUNK * BLOCK) * 8u;
#pragma unroll
        for (int tc = 0; tc < CHUNK; ++tc) {
            int t = c * CHUNK + tc; if (t > T - 1) t = T - 1;   // clamp tail
            async_copy_f2(sb + (uint32_t)(tc * BLOCK + tid) * 8u, pe,
                          ((uint32_t)t * (uint32_t)B + (uint32_t)bc) * 8u);
        }
    }

    // ---- state (all zeros) ----
    float auztw = 0.f, alztw = 0.f, uztw = 0.f, uzfw = 0.f, lztw = 0.f;
    float lzfs = 0.f, lzfp = 0.f, qi0 = 0.f, qgs0 = 0.f, qgp0 = 0.f;

    float2* outv = (float2*)out;

    for (int c = 0; c < nchunk; ++c) {
        // chunks still possibly in flight beyond c (compile-time immediates only)
        const int issued = (c + DEPTH < nchunk) ? (c + DEPTH) : nchunk;
        const int beyond = issued - c - 1;
        if (beyond >= 2)      asm volatile("s_wait_asynccnt 32" ::: "memory");
        else if (beyond == 1) asm volatile("s_wait_asynccnt 16" ::: "memory");
        else                  asm volatile("s_wait_asynccnt 0"  ::: "memory");

        const int slot = c % DEPTH;
        const float2* sbuf = &buf[slot * CHUNK * BLOCK];
        const int tmax = T - c * CHUNK;

#pragma unroll 1
        for (int tc = 0; tc < CHUNK; ++tc) {
            if (tc >= tmax) break;
            const float2 v = sbuf[tc * BLOCK + tid];          // ds_load_b64

            const float prcp = fmaxf(v.x, 0.0f);
            const float pet  = fmaxf(v.y, 0.0f);
            const float ep   = kc * pet;

            // impervious / alterable-impervious area
            const float roimp = prcp * pctim;
            const float ae2   = pctim * ep;
            const float ae1   = fminf(auztw, ep * auztw * inv_uztwm);
            const float ae3   = (ep - ae1) * alztw * inv_ul;
            const float pav   = fmaxf(0.0f, prcp - (uztwm - (auztw - ae1)));
            const float alzt  = alztw - ae3;
            const float adsur = pav * alzt * inv_lztwm;
            const float ars   = fmaxf(0.0f, pav - adsur + alzt - lztwm);
            const float auztw_n = fminf(uztwm, auztw - ae1 + prcp);
            const float alztw_n = fminf(lztwm, pav - adsur + alzt);

            // pervious-area evaporation
            const float e1 = fminf(uztw, ep * uztw * inv_uztwm);
            const float e2 = fminf(uzfw, ep - e1);
            const float e3 = (ep - e1 - e2) * lztw * inv_ul;
            float lt = lztw - e3;
            const float e4 = riva * ep;
            const float e_sim = ae2 + ae1 + ae3 + e1 + e2 + e3 + e4;

            // surface runoff + upper-layer update
            const float uzsum = uztw + uzfw - e1 - e2;
            const float rs = fmaxf(prcp + uzsum - (uztwm + uzfwm), 0.0f) * parea;
            const float ut = fminf(uztwm, uztw - e1 + prcp);
            float uf = fminf(uzfwm, prcp + uzsum - ut);
            const float ri = uf * uzk;
            uf -= ri;

            // percolation (fast pow: exp2(rexp * log2(defr)))
            const float lsum = lzfs + lzfp + lt;
            const float defr = fmaxf(1.0f - lsum * inv_cap_l, 1e-6f);
            const float dpow = __builtin_amdgcn_exp2f(rexp * __builtin_amdgcn_logf(defr));
            const float perc = pbase * (1.0f + zperc * dpow) * uf * inv_uzfwm;
            const float rate = fminf(perc, cap_l - lsum);
            const float fx   = fminf(lzfsum - (lzfs + lzfp),
                                     fmaxf(rate - (lztwm - lt), rate * pfree));
            const float perct = rate - fx;
            const float x     = 1.0f - lzfp * inv_lzfpm;   // verbatim (2x)/(x + 0)
            const float coef  = fminf(coef_a * (2.0f * x) / x, 1.0f);
            const float percp = fminf(lzfpm - lzfp,
                                      fmaxf(fx - (lzfsm - lzfs), coef * fx));
            const float percs = fx - percp;
            lt += perct;
            float ls = lzfs + percs;
            float lp = lzfp + percp;
            const float rgs = ls * lzsk; ls -= rgs;
            const float rgp = lp * lzpk; lp -= rgp;

            // upper-layer redistribution
            const bool  c1  = (ut * inv_uztwm) < (uf * inv_uzfwm);
            const float mix = (ut + uf) * inv_uu;
            const float uztw_n = c1 ? uztwm * mix : ut;
            const float uzfw_n = c1 ? uzfwm * mix : uf;

            // lower-layer redistribution
            const float ratio = fmaxf((ls + lp - saved + lt) * inv_rden, 0.0f);
            const bool  c2   = (lt * inv_lztwm) < ratio;
            const float del  = lztwm * ratio - lt;
            const float lztw_n = c2 ? lztwm * ratio : lt;
            const float lzfs_n = c2 ? fmaxf(0.0f, ls - del) : ls;
            const float lzfp_n = c2 ? (lp - fmaxf(0.0f, del - ls)) : lp;

            // routing
            const float rs_tot = roimp + (adsur + ars) * adimp + rs;
            const float qs  = rs_tot * 1000000.0f;   // * AREA * 1000
            const float qi  = ci  * qi0  + kiu  * ri;
            const float qgs = cgs * qgs0 + kgsu * rgs;
            const float qgp = cgp * qgp0 + kgpu * rgp;
            const float q_sim = qs + qi + qgs + qgp;

            // carry state
            auztw = auztw_n; alztw = alztw_n; uztw = uztw_n; uzfw = uzfw_n;
            lztw = lztw_n; lzfs = lzfs_n; lzfp = lzfp_n;
            qi0 = qi; qgs0 = qgs; qgp0 = qgp;

            if (live)
                outv[(size_t)(c * CHUNK + tc) * (size_t)B + (size_t)bc] =
                    make_float2(q_sim, e_sim);
        }

        // refill the slot we just drained
        const int ic = c + DEPTH;
        if (ic < nchunk) {
            const uint32_t sb = lds0 + (uint32_t)(slot * CHUNK * BLOCK) * 8u;
#pragma unroll
            for (int tc = 0; tc < CHUNK; ++tc) {
                int t = ic * CHUNK + tc; if (t > T - 1) t = T - 1;
                async_copy_f2(sb + (uint32_t)(tc * BLOCK + tid) * 8u, pe,
                              ((uint32_t)t * (uint32_t)B + (uint32_t)bc) * 8u);
            }
        }
    }
    // outstanding async loads drained by s_endpgm's implicit wait-idle
}

extern "C" void kernel_launch(void* const* d_in, const int* in_sizes, int n_in,
                              void* d_out, int out_size, void* d_ws, size_t ws_size,
                              hipStream_t stream) {
    const float* pe  = (const float*)d_in[0];   // [T, B, 2]
    const float* prm = (const float*)d_in[1];   // [B, 21]
    float* out = (float*)d_out;                 // [T, B, 2]

    const int B = in_sizes[1] / 21;
    const int T = in_sizes[0] / (2 * B);

    dim3 grid((B + BLOCK - 1) / BLOCK), block(BLOCK);
    sac_sma_kernel<<<grid, block, 0, stream>>>(pe, prm, out, T, B);
}